// NoisyModalRouter_2D_24206435680554
// MI455X (gfx1250) — compile-verified
//
#include <hip/hip_runtime.h>
#include <math.h>

// CDNA5 / gfx1250: wave32, V_WMMA_F32_16X16X4_F32 + Tensor Data Mover.
typedef __attribute__((ext_vector_type(2))) float v2f;
typedef __attribute__((ext_vector_type(8))) float v8f;
typedef __attribute__((ext_vector_type(4))) unsigned u32x4;
typedef __attribute__((ext_vector_type(8))) int i32x8;
typedef __attribute__((ext_vector_type(4))) int i32x4;

#define M_  4
#define B_  8
#define C_  256
#define HW_ 16384          // 128*128
#define TP_ 128            // tile pixels per block (contiguous, = 512B rows)
#define TC_ 32             // tile channels per chunk
#define NCHUNK (C_ / TC_)  // 8 chunks per modal
#define NIT (M_ * NCHUNK)  // 32 tiles per block

#if defined(__has_builtin)
#if __has_builtin(__builtin_amdgcn_tensor_load_to_lds)
#define HAVE_TDM 1
#endif
#endif

// ---- deterministic counter-based RNG (PCG hash + Box-Muller) --------------
__device__ __forceinline__ unsigned pcg_hash(unsigned v) {
  unsigned s = v * 747796405u + 2891336453u;
  unsigned w = ((s >> ((s >> 28) + 4u)) ^ s) * 277803737u;
  return (w >> 22) ^ w;
}

__device__ __forceinline__ float gauss(unsigned idx) {
  unsigned ha = pcg_hash(idx);
  unsigned hb = pcg_hash(idx ^ 0x9e3779b9u);
  float u1 = ((float)ha + 1.0f) * 2.3283064365386963e-10f; // (0,1]
  float u2 = (float)hb * 2.3283064365386963e-10f;          // [0,1)
  float r = sqrtf(-2.0f * logf(u1));
  return r * __cosf(6.283185307179586f * u2);
}

__device__ __forceinline__ float softplus_f(float x) {
  return (x > 20.0f) ? x : log1pf(expf(x));
}

#if HAVE_TDM
// Issue one TDM 2D tile load: TC_ rows (channel dim, stride HW_) x TP_
// pixels (contiguous 512B rows) of fp32, global -> LDS. D# per ISA §8.
__device__ __forceinline__ void tdm_issue(const float* gsrc,
                                          unsigned lds_byte_off) {
  const unsigned long long ga = (unsigned long long)(uintptr_t)gsrc;
  // Group 0: count=1 | lds_addr | global_addr[56:0] | type=2 ("image")
  u32x4 g0 = {1u, lds_byte_off, (unsigned)(ga & 0xffffffffu),
              (unsigned)((ga >> 32) & 0x01ffffffu) | (2u << 30)};
  // Group 1: data_size=2 (4B); tensor_dim0=HW_, tensor_dim1=C_;
  //          tile_dim0=TP_, tile_dim1=TC_, tile_dim2=0;
  //          tensor_dim0_stride=HW_ (row-to-row channel stride).
  i32x8 g1 = {
      (int)0x00020000u,                                   // data_size=4B
      (int)((HW_ & 0xffff) << 16),                        // tdim0[15:0]
      (int)(((HW_ >> 16) & 0xffff) | ((C_ & 0xffff) << 16)),  // tdim0hi|tdim1lo
      (int)(((C_ >> 16) & 0xffff) | (TP_ << 16)),         // tdim1hi|tile_dim0
      (int)(TC_ & 0xffff),                                // tile_dim1|tile_dim2=0
      (int)HW_,                                           // tdim0_stride[31:0]
      0, 0};                                              // stride hi / dim1 stride
  i32x4 gz4 = {0, 0, 0, 0};        // 2D tensor: groups 2/3 unused
  i32x8 gz8 = {0, 0, 0, 0, 0, 0, 0, 0};
#if __clang_major__ >= 23
  // amdgpu-toolchain / therock-10.0 headers: 6-arg form.
  __builtin_amdgcn_tensor_load_to_lds(g0, g1, gz4, gz4, gz8, 0);
#else
  // ROCm 7.2 (clang-22): 5-arg form.
  __builtin_amdgcn_tensor_load_to_lds(g0, g1, gz4, gz4, 0);
#endif
}
#endif

// ---------------------------------------------------------------------------
// Block = 256 threads (8 wave32) -> 128 consecutive pixels of one (b) slab.
// TDM streams [32 channels x 128 pixels] fp32 tiles into a double-buffered
// LDS stage; each wave runs WMMA (16 pixels x 4K per op) from LDS.
// B-matrix col0 = score_w[m], col1 = noise_w[m], cols>=2 read a zero row so
// EXEC stays all-ones for WMMA. Softmax over the 4 modals is done in-block.
// ---------------------------------------------------------------------------
__global__ __launch_bounds__(256) void noisy_modal_router_kernel(
    const float* __restrict__ feats,    // [M,B,C,H,W]
    const float* __restrict__ score_w,  // [M,C]
    const float* __restrict__ noise_w,  // [M,C]
    float* __restrict__ out) {          // [M,B,1,H,W]
  __shared__ float lds_tile[2][TC_ * TP_];   // 2 x 16 KiB stream buffers
  __shared__ float lds_w[M_ * 2 * C_];       // [m][sel][c]
  __shared__ float lds_zero[C_];             // zero column source (N>=2)
  __shared__ float lds_res[8][M_][16][2];    // [wave][m][pixel][score|std]

  const int tid  = threadIdx.x;
  const int lane = tid & 31;
  const int wave = tid >> 5;

  // Stage weights into LDS: i = m*512 + sel*256 + c.
  for (int i = tid; i < M_ * 2 * C_; i += 256) {
    const int m = i >> 9, sel = (i >> 8) & 1, c = i & 255;
    lds_w[i] = sel ? noise_w[m * C_ + c] : score_w[m * C_ + c];
  }
  lds_zero[tid] = 0.0f;
  __syncthreads();

  const int b  = blockIdx.x >> 7;           // 128 blocks per batch slab
  const int p0 = (blockIdx.x & 127) << 7;   // 128-pixel tile origin

  // WMMA fp32 A layout: lanes 0-15 row=lane K={0,1}; lanes 16-31 K={2,3}.
  const int arow = lane & 15;
  const int koff = (lane >> 4) << 1;
  const int col  = lane & 15;
  const int pix  = (wave << 4) + arow;      // pixel column in LDS tile

#if HAVE_TDM
  const unsigned toff0 = (unsigned)(uintptr_t)&lds_tile[0][0];
  const unsigned toff1 = (unsigned)(uintptr_t)&lds_tile[1][0];
  if (wave == 0)
    tdm_issue(feats + ((size_t)b * C_) * HW_ + p0, toff0);  // m=0, ck=0
#endif

  const v8f vzero = {};
  v8f acc = vzero;

  for (int it = 0; it < NIT; ++it) {
    const int m = it >> 3, ck = it & 7;

#if HAVE_TDM
    if (wave == 0) {
      if (it + 1 < NIT) {
        const int m2 = (it + 1) >> 3, ck2 = (it + 1) & 7;
        tdm_issue(feats + ((size_t)(m2 * B_ + b) * C_ + (size_t)(ck2 * TC_)) *
                              HW_ + p0,
                  ((it + 1) & 1) ? toff1 : toff0);
        __builtin_amdgcn_s_wait_tensorcnt(1);  // tile `it` landed
      } else {
        __builtin_amdgcn_s_wait_tensorcnt(0);
      }
    }
    __syncthreads();  // tile `it` visible to all waves
#else
    // Fallback: cooperative synchronous fill of buf[it&1].
    {
      float* dst = &lds_tile[it & 1][0];
      const float* src =
          feats + ((size_t)(m * B_ + b) * C_ + (size_t)(ck * TC_)) * HW_ + p0;
      for (int i = tid; i < TC_ * TP_; i += 256)
        dst[i] = src[(size_t)(i >> 7) * HW_ + (i & 127)];
    }
    __syncthreads();
#endif

    const float* tile = &lds_tile[it & 1][0];
    const float* wp = (col == 0)   ? &lds_w[(m * 2 + 0) * C_]
                      : (col == 1) ? &lds_w[(m * 2 + 1) * C_]
                                   : lds_zero;
#pragma unroll
    for (int kk = 0; kk < TC_ / 4; ++kk) {
      const int cl = (kk << 2) + koff;   // channel within chunk
      const int cg = (ck << 5) + cl;     // channel within C for weights
      v2f a, bv;
      a.x  = tile[cl * TP_ + pix];       // conflict-free: 64B rows per half-wave
      a.y  = tile[(cl + 1) * TP_ + pix];
      bv.x = wp[cg];
      bv.y = wp[cg + 1];
      acc = __builtin_amdgcn_wmma_f32_16x16x4_f32(
          false, a, false, bv, (short)0, acc, false, false);
    }

    if (ck == 7) {  // modal finished: scatter cols 0/1 of D, reset acc
      if (col < 2) {
        const int rbase = (lane >> 4) << 3;
#pragma unroll
        for (int r = 0; r < 8; ++r)
          lds_res[wave][m][rbase + r][col] = acc[r];
      }
      acc = vzero;
    }
    __syncthreads();  // safe to overwrite buf[it&1] next round
  }

  // Lanes 0-15 finalize one pixel each: noise, softplus, softmax over m.
  if (lane < 16) {
    const int hw = p0 + (wave << 4) + lane;
    float logit[M_];
#pragma unroll
    for (int m = 0; m < M_; ++m) {
      const float score = lds_res[wave][m][lane][0];
      const float sraw  = lds_res[wave][m][lane][1];
      const unsigned idx =
          (unsigned)((m * B_ + b) * HW_ + hw) * 2654435761u + 42u;
      logit[m] = score + gauss(idx) * softplus_f(sraw);
    }
    const float mx =
        fmaxf(fmaxf(logit[0], logit[1]), fmaxf(logit[2], logit[3]));
    float e[M_];
    float sum = 0.0f;
#pragma unroll
    for (int m = 0; m < M_; ++m) {
      e[m] = expf(logit[m] - mx);
      sum += e[m];
    }
    const float inv = 1.0f / sum;
#pragma unroll
    for (int m = 0; m < M_; ++m)
      out[(size_t)(m * B_ + b) * HW_ + hw] = e[m] * inv;
  }
}

extern "C" void kernel_launch(void* const* d_in, const int* in_sizes, int n_in,
                              void* d_out, int out_size, void* d_ws,
                              size_t ws_size, hipStream_t stream) {
  (void)in_sizes; (void)n_in; (void)out_size; (void)d_ws; (void)ws_size;
  const float* feats   = (const float*)d_in[0];
  const float* score_w = (const float*)d_in[1];
  const float* noise_w = (const float*)d_in[2];
  float* out = (float*)d_out;

  // 1024 blocks: 8 batch slabs x 128 pixel-tiles, 256 threads (8 wave32).
  noisy_modal_router_kernel<<<dim3(1024), dim3(256), 0, stream>>>(
      feats, score_w, noise_w, out);
}